// StructuralAttentionLayer_10608569221674
// MI455X (gfx1250) — compile-verified
//
#include <hip/hip_runtime.h>
#include <hip/hip_bf16.h>
#include <math.h>

typedef __attribute__((ext_vector_type(2))) float v2f;
typedef __attribute__((ext_vector_type(8))) float v8f;

#define D_IN   256
#define OUT_SZ 32
#define NHEAD  4
#define FTS_W  128   // NHEAD * OUT_SZ
#define ALPHA  0.2f

// ---- float <-> total-order uint encoding (for atomic max on floats) ----
__device__ __forceinline__ unsigned ord_encode(float f) {
    unsigned u = __float_as_uint(f);
    return (u & 0x80000000u) ? ~u : (u | 0x80000000u);
}
__device__ __forceinline__ float ord_decode(unsigned e) {
    unsigned u = (e & 0x80000000u) ? (e ^ 0x80000000u) : ~e;
    return __uint_as_float(u);
}

// ---- K0: zero output, init segmax/denom ----
__global__ void gat_init(float* __restrict__ out, unsigned* __restrict__ segmax,
                         float* __restrict__ denom, int outN, int hnN) {
    int t = blockIdx.x * blockDim.x + threadIdx.x;
    if (t < outN) out[t] = 0.0f;
    if (t < hnN) { segmax[t] = 0u; denom[t] = 0.0f; }  // 0u encodes < any finite float
}

// ---- K1: fts[N][128] = x[N][256] @ W  via V_WMMA_F32_16X16X4_F32 ----
// One wave computes a full 16x128 row-slab: A fragment loaded once per K-step,
// reused by 8 WMMAs (one per 16-col tile). W layout: [H][D_IN][OUT_SZ].
__global__ void gat_gemm(const float* __restrict__ x, const float* __restrict__ W,
                         float* __restrict__ fts, int Nn) {
    const int lane = threadIdx.x & 31;
    const int tm   = blockIdx.x * (blockDim.x >> 5) + (threadIdx.x >> 5);
    if (tm * 16 >= Nn) return;                    // wave-uniform guard

    int ml = tm * 16 + (lane & 15);
    if (ml >= Nn) ml = Nn - 1;                    // clamp loads, keep EXEC full
    const int kk = (lane < 16) ? 0 : 2;           // K-split across lane halves

    const float* __restrict__ xrow = x + (size_t)ml * D_IN;

    // per-N-tile B column pointers: col n = t*16 + (lane&15); W[h*8192 + d*32 + o]
    const float* wcol[8];
    #pragma unroll
    for (int t = 0; t < 8; ++t) {
        const int n = t * 16 + (lane & 15);
        wcol[t] = W + (size_t)(n >> 5) * (D_IN * OUT_SZ) + (n & 31);
    }

    v8f acc[8] = {};
    for (int k = 0; k < D_IN; k += 4) {
        v2f a;
        a.x = xrow[k + kk];
        a.y = xrow[k + kk + 1];
        #pragma unroll
        for (int t = 0; t < 8; ++t) {
            v2f b;
            b.x = wcol[t][(size_t)(k + kk) * OUT_SZ];
            b.y = wcol[t][(size_t)(k + kk + 1) * OUT_SZ];
            acc[t] = __builtin_amdgcn_wmma_f32_16x16x4_f32(false, a, false, b,
                                                           (short)0, acc[t], false, false);
        }
    }

    // C/D layout: VGPR i -> row (i | (lane<16?0:8)), col = lane&15
    const int row0 = tm * 16 + ((lane < 16) ? 0 : 8);
    const bool full = (tm * 16 + 16 <= Nn);       // wave-uniform
    if (full) {
        #pragma unroll
        for (int t = 0; t < 8; ++t) {
            float* __restrict__ op = fts + (size_t)row0 * FTS_W + t * 16 + (lane & 15);
            #pragma unroll
            for (int i = 0; i < 8; ++i) op[(size_t)i * FTS_W] = acc[t][i];
        }
    } else {
        #pragma unroll
        for (int t = 0; t < 8; ++t) {
            float* __restrict__ op = fts + (size_t)row0 * FTS_W + t * 16 + (lane & 15);
            #pragma unroll
            for (int i = 0; i < 8; ++i)
                if (row0 + i < Nn) op[(size_t)i * FTS_W] = acc[t][i];
        }
    }
}

// ---- K2: f1/f2[n][h] = <fts[n,h,:], a{1,2}[h,:]> + b{1,2}[h] ----
__global__ void gat_proj(const float* __restrict__ fts,
                         const float* __restrict__ a1, const float* __restrict__ b1,
                         const float* __restrict__ a2, const float* __restrict__ b2,
                         float* __restrict__ f1, float* __restrict__ f2, int Nn) {
    int t = blockIdx.x * blockDim.x + threadIdx.x;
    if (t >= Nn * NHEAD) return;
    const int n = t >> 2, h = t & 3;
    const float* __restrict__ p  = fts + (size_t)n * FTS_W + h * OUT_SZ;
    const float* __restrict__ A1 = a1 + h * OUT_SZ;
    const float* __restrict__ A2 = a2 + h * OUT_SZ;
    float s1 = 0.0f, s2 = 0.0f;
    #pragma unroll
    for (int o = 0; o < OUT_SZ; ++o) {
        float v = p[o];
        s1 += v * A1[o];
        s2 += v * A2[o];
    }
    f1[n * NHEAD + h] = s1 + b1[h];
    f2[n * NHEAD + h] = s2 + b2[h];
}

// ---- K3: per-edge leaky-relu logits, atomic segment max over rows ----
__global__ void gat_segmax(const int* __restrict__ erow, const int* __restrict__ ecol,
                           const float* __restrict__ f1, const float* __restrict__ f2,
                           unsigned* __restrict__ segmax, int En) {
    int e = blockIdx.x * blockDim.x + threadIdx.x;
    if (e >= En) return;
    const int r = erow[e], c = ecol[e];
    const float4 v1 = *(const float4*)(f1 + (size_t)r * NHEAD);
    const float4 v2 = *(const float4*)(f2 + (size_t)c * NHEAD);
    float lr[4] = { v1.x + v2.x, v1.y + v2.y, v1.z + v2.z, v1.w + v2.w };
    #pragma unroll
    for (int h = 0; h < NHEAD; ++h) {
        float l = lr[h];
        l = fmaxf(ALPHA * l, l);
        atomicMax(&segmax[(size_t)r * NHEAD + h], ord_encode(l));
    }
}

// ---- K4: denom[r][h] += exp(lr - segmax[r][h]) ----
__global__ void gat_segsum(const int* __restrict__ erow, const int* __restrict__ ecol,
                           const float* __restrict__ f1, const float* __restrict__ f2,
                           const unsigned* __restrict__ segmax, float* __restrict__ denom,
                           int En) {
    int e = blockIdx.x * blockDim.x + threadIdx.x;
    if (e >= En) return;
    const int r = erow[e], c = ecol[e];
    const float4 v1 = *(const float4*)(f1 + (size_t)r * NHEAD);
    const float4 v2 = *(const float4*)(f2 + (size_t)c * NHEAD);
    float lr[4] = { v1.x + v2.x, v1.y + v2.y, v1.z + v2.z, v1.w + v2.w };
    #pragma unroll
    for (int h = 0; h < NHEAD; ++h) {
        float l = lr[h];
        l = fmaxf(ALPHA * l, l);
        float ex = expf(l - ord_decode(segmax[(size_t)r * NHEAD + h]));
        atomicAdd(&denom[(size_t)r * NHEAD + h], ex);
    }
}

// ---- K5: aggregation. One wave per edge; 32 lanes x float4 cover 128 channels ----
__global__ void gat_aggregate(const int* __restrict__ erow, const int* __restrict__ ecol,
                              const float* __restrict__ f1, const float* __restrict__ f2,
                              const unsigned* __restrict__ segmax,
                              const float* __restrict__ denom,
                              const float* __restrict__ fts,
                              float* __restrict__ out, int En) {
    const int lane = threadIdx.x & 31;
    const int e = blockIdx.x * (blockDim.x >> 5) + (threadIdx.x >> 5);
    if (e >= En) return;
    const int r = erow[e], c = ecol[e];
    const int h = lane >> 3;                       // lane covers channels lane*4..lane*4+3
    float l = f1[(size_t)r * NHEAD + h] + f2[(size_t)c * NHEAD + h];
    l = fmaxf(ALPHA * l, l);
    const float m = ord_decode(segmax[(size_t)r * NHEAD + h]);
    const float coef = expf(l - m) / denom[(size_t)r * NHEAD + h];
    const float4 fv = *(const float4*)(fts + (size_t)c * FTS_W + lane * 4);
    float* __restrict__ op = out + (size_t)r * FTS_W + lane * 4;
    atomicAdd(op + 0, coef * fv.x);
    atomicAdd(op + 1, coef * fv.y);
    atomicAdd(op + 2, coef * fv.z);
    atomicAdd(op + 3, coef * fv.w);
}

// ---- K6: ELU in place ----
__global__ void gat_elu(float* __restrict__ out, int n) {
    int t = blockIdx.x * blockDim.x + threadIdx.x;
    if (t >= n) return;
    float v = out[t];
    out[t] = (v > 0.0f) ? v : (expf(v) - 1.0f);
}

extern "C" void kernel_launch(void* const* d_in, const int* in_sizes, int n_in,
                              void* d_out, int out_size, void* d_ws, size_t ws_size,
                              hipStream_t stream) {
    const float* x    = (const float*)d_in[0];
    const float* W    = (const float*)d_in[1];
    const float* a1   = (const float*)d_in[2];
    const float* b1   = (const float*)d_in[3];
    const float* a2   = (const float*)d_in[4];
    const float* b2   = (const float*)d_in[5];
    const int*   erow = (const int*)d_in[6];
    const int*   ecol = (const int*)d_in[7];
    float* out = (float*)d_out;

    const int Nn = in_sizes[0] / D_IN;   // 100000
    const int En = in_sizes[6];          // 1600000

    // workspace layout
    float*    fts    = (float*)d_ws;                          // N*128
    float*    f1     = fts + (size_t)Nn * FTS_W;              // N*4
    float*    f2     = f1 + (size_t)Nn * NHEAD;               // N*4
    unsigned* segmax = (unsigned*)(f2 + (size_t)Nn * NHEAD);  // N*4
    float*    denom  = (float*)(segmax + (size_t)Nn * NHEAD); // N*4

    const int outN = Nn * FTS_W;
    const int hnN  = Nn * NHEAD;

    // K0: init
    gat_init<<<(outN + 255) / 256, 256, 0, stream>>>(out, segmax, denom, outN, hnN);

    // K1: WMMA GEMM — one wave per 16-row slab (computes all 8 col-tiles)
    {
        int waves  = (Nn + 15) / 16;
        int blocks = (waves + 3) / 4;
        gat_gemm<<<blocks, 128, 0, stream>>>(x, W, fts, Nn);
    }
    // K2: projections
    gat_proj<<<(Nn * NHEAD + 255) / 256, 256, 0, stream>>>(fts, a1, b1, a2, b2, f1, f2, Nn);
    // K3: segment max
    gat_segmax<<<(En + 255) / 256, 256, 0, stream>>>(erow, ecol, f1, f2, segmax, En);
    // K4: segment sum of exp
    gat_segsum<<<(En + 255) / 256, 256, 0, stream>>>(erow, ecol, f1, f2, segmax, denom, En);
    // K5: aggregation, one wave per edge (8 waves per 256-thread block)
    gat_aggregate<<<(En + 7) / 8, 256, 0, stream>>>(erow, ecol, f1, f2, segmax, denom, fts, out, En);
    // K6: ELU
    gat_elu<<<(outN + 255) / 256, 256, 0, stream>>>(out, outN);
}